// CGCNNEncoder_40965398069686
// MI455X (gfx1250) — compile-verified
//
#include <hip/hip_runtime.h>
#include <math.h>

#define GNUM   128
#define APG    32
#define NATOMS 4096
#define KNN    12
#define EDGES  (NATOMS*KNN)   // 49152
#define AF     64
#define NF     64
#define FIN    192            // 2*AF + NF
#define C2     128            // 2*AF
#define LD     128
#define WSTRIDE 194           // padded LDS row stride (dwords), 8B-aligned rows

typedef __attribute__((ext_vector_type(2))) float v2f;
typedef __attribute__((ext_vector_type(8))) float v8f;

// ---------------- cart coords: cart[n][j] = sum_i fracs[n][i] * lat[b][i][j] -----------
__global__ void cgc_cart(const float* __restrict__ fracs, const float* __restrict__ lat,
                         float* __restrict__ cart) {
  int n = blockIdx.x * blockDim.x + threadIdx.x;
  if (n >= NATOMS) return;
  int b = n / APG;
  float f0 = fracs[n*3+0], f1 = fracs[n*3+1], f2 = fracs[n*3+2];
  const float* L = lat + b*9;
#pragma unroll
  for (int j = 0; j < 3; ++j)
    cart[n*3+j] = f0*L[j] + f1*L[3+j] + f2*L[6+j];
}

// ---------------- per-graph kNN (one wave per graph of 32 atoms) -----------------------
__global__ void cgc_knn(const float* __restrict__ cart, int* __restrict__ esrc,
                        float* __restrict__ edist) {
  __shared__ float sc[APG*3];
  __shared__ float sd[APG][APG];
  int g = blockIdx.x, a = threadIdx.x;          // a in [0,32)
  int gi = g*APG + a;
  sc[a*3+0] = cart[gi*3+0];
  sc[a*3+1] = cart[gi*3+1];
  sc[a*3+2] = cart[gi*3+2];
  __syncthreads();
  float ax = sc[a*3], ay = sc[a*3+1], az = sc[a*3+2];
  for (int j = 0; j < APG; ++j) {
    float dx = ax - sc[j*3], dy = ay - sc[j*3+1], dz = az - sc[j*3+2];
    sd[a][j] = (j == a) ? 1e30f : (dx*dx + dy*dy + dz*dz);
  }
  // stable selection of 12 smallest (first index wins ties, like jax top_k)
  for (int p = 0; p < KNN; ++p) {
    float best = 1e30f; int bj = 0;
    for (int j = 0; j < APG; ++j) {
      float d = sd[a][j];
      if (d < best) { best = d; bj = j; }
    }
    sd[a][bj] = 1e30f;
    esrc [gi*KNN + p] = g*APG + bj;
    edist[gi*KNN + p] = sqrtf(best + 1e-8f);
  }
}

// ---------------- atom feature init: emb[species] ------------------------------------
__global__ void cgc_fea_init(const int* __restrict__ species, const float* __restrict__ emb,
                             float* __restrict__ fea) {
  int idx = blockIdx.x * blockDim.x + threadIdx.x;   // N*AF threads
  int n = idx >> 6, c = idx & 63;
  fea[idx] = emb[species[n]*AF + c];
}

// ---------------- edge GEMM: total[E,128] = [src|dst|nbr][E,192] @ W[192,128] + b -----
// 4 waves / block; wave = 16 edge rows x full 128 cols; f32 WMMA 16x16x4.
__global__ __launch_bounds__(128) void cgc_edge_gemm(
    const float* __restrict__ fea, const int* __restrict__ esrc,
    const float* __restrict__ edist, const float* __restrict__ W,
    const float* __restrict__ bias, float* __restrict__ total) {
  extern __shared__ float Wt[];                      // [128][WSTRIDE], transposed W
  int tid = threadIdx.x;
  for (int idx = tid; idx < FIN*C2; idx += 128) {
    int k = idx >> 7, n = idx & 127;                 // W row-major [k][n]
    Wt[n*WSTRIDE + k] = W[idx];
  }
  __syncthreads();

  int wave = tid >> 5, lane = tid & 31;
  int hf   = lane >> 4;                              // lane half selects K-pair
  int lm   = lane & 15;                              // A: row M ; B/C: col N
  int kp   = hf * 2;
  int row  = blockIdx.x*64 + wave*16 + lm;           // edge row this lane feeds (A operand)
  int s    = esrc[row];
  int dd   = row / KNN;                              // edge_dst = repeat(arange(N), K)
  float dist = edist[row];
  const float STEP = 8.0f / 63.0f;
  const float CO   = -0.5f / (STEP * STEP);

  v8f acc[8];
#pragma unroll
  for (int t = 0; t < 8; ++t)
#pragma unroll
    for (int r = 0; r < 8; ++r) acc[t][r] = 0.0f;

  const float* srow = fea + s  * AF;
  const float* drow = fea + dd * AF;

  // k in [0,64): neighbor(src) features
  for (int kb = 0; kb < 64; kb += 4) {
    float2 av = *(const float2*)(srow + kb + kp);
    v2f a; a[0] = av.x; a[1] = av.y;
#pragma unroll
    for (int t = 0; t < 8; ++t) {
      int n = t*16 + lm;
      float2 bv = *(const float2*)(&Wt[n*WSTRIDE + kb + kp]);
      v2f b; b[0] = bv.x; b[1] = bv.y;
      acc[t] = __builtin_amdgcn_wmma_f32_16x16x4_f32(false, a, false, b, (short)0, acc[t], false, false);
    }
  }
  // k in [64,128): center(dst) features
  for (int kb = 0; kb < 64; kb += 4) {
    float2 av = *(const float2*)(drow + kb + kp);
    v2f a; a[0] = av.x; a[1] = av.y;
#pragma unroll
    for (int t = 0; t < 8; ++t) {
      int n = t*16 + lm;
      float2 bv = *(const float2*)(&Wt[n*WSTRIDE + 64 + kb + kp]);
      v2f b; b[0] = bv.x; b[1] = bv.y;
      acc[t] = __builtin_amdgcn_wmma_f32_16x16x4_f32(false, a, false, b, (short)0, acc[t], false, false);
    }
  }
  // k in [128,192): Gaussian smearing computed on the fly
  for (int kb = 0; kb < 64; kb += 4) {
    int k0 = kb + kp;
    float t0 = dist - (float)k0       * STEP;
    float t1 = dist - (float)(k0 + 1) * STEP;
    v2f a; a[0] = expf(CO*t0*t0); a[1] = expf(CO*t1*t1);
#pragma unroll
    for (int t = 0; t < 8; ++t) {
      int n = t*16 + lm;
      float2 bv = *(const float2*)(&Wt[n*WSTRIDE + 128 + kb + kp]);
      v2f b; b[0] = bv.x; b[1] = bv.y;
      acc[t] = __builtin_amdgcn_wmma_f32_16x16x4_f32(false, a, false, b, (short)0, acc[t], false, false);
    }
  }

  // epilogue: C/D layout -> row = base + r + 8*hf, col = t*16 + lm
  int orow = blockIdx.x*64 + wave*16 + 8*hf;
#pragma unroll
  for (int t = 0; t < 8; ++t) {
    int n = t*16 + lm;
    float bv = bias[n];
#pragma unroll
    for (int r = 0; r < 8; ++r)
      total[(size_t)(orow + r)*C2 + n] = acc[t][r] + bv;
  }
}

// ---------------- column mean/var (biased) over `rows` rows --------------------------
__global__ void cgc_col_stats(const float* __restrict__ x, int rows, int cols,
                              float* __restrict__ mean, float* __restrict__ var) {
  __shared__ float ss[256], sq[256];
  int c = blockIdx.x, t = threadIdx.x;
  float s = 0.0f, q = 0.0f;
  for (int r = t; r < rows; r += 256) {
    float v = x[(size_t)r*cols + c];
    s += v; q += v*v;
  }
  ss[t] = s; sq[t] = q;
  __syncthreads();
  for (int off = 128; off > 0; off >>= 1) {
    if (t < off) { ss[t] += ss[t+off]; sq[t] += sq[t+off]; }
    __syncthreads();
  }
  if (t == 0) {
    float m = ss[0] / (float)rows;
    mean[c] = m;
    var[c]  = sq[0] / (float)rows - m*m;
  }
}

// ---------------- BN1 + sigmoid*softplus + segment-sum over 12 edges -----------------
__global__ void cgc_msg_seg(const float* __restrict__ total, const float* __restrict__ mean,
                            const float* __restrict__ var, const float* __restrict__ g1,
                            const float* __restrict__ bt1, float* __restrict__ upd) {
  int idx = blockIdx.x * blockDim.x + threadIdx.x;   // N*AF threads
  int i = idx >> 6, c = idx & 63;
  float m1 = mean[c],    is1 = rsqrtf(var[c]    + 1e-5f), ga = g1[c],    ba = bt1[c];
  float m2 = mean[c+64], is2 = rsqrtf(var[c+64] + 1e-5f), gb = g1[c+64], bb = bt1[c+64];
  float acc = 0.0f;
  for (int j = 0; j < KNN; ++j) {
    size_t r = (size_t)(i*KNN + j) * C2;
    float xf = ga*(total[r + c]      - m1)*is1 + ba;
    float xc = gb*(total[r + c + 64] - m2)*is2 + bb;
    float sig = 1.0f / (1.0f + expf(-xf));
    float sp  = fmaxf(xc, 0.0f) + log1pf(expf(-fabsf(xc)));
    acc += sig * sp;
  }
  upd[idx] = acc;
}

// ---------------- atom update: fea = softplus(fea + BN2(upd)) ------------------------
__global__ void cgc_fea_update(float* __restrict__ fea, const float* __restrict__ upd,
                               const float* __restrict__ mean, const float* __restrict__ var,
                               const float* __restrict__ g2, const float* __restrict__ bt2) {
  int idx = blockIdx.x * blockDim.x + threadIdx.x;   // N*AF threads
  int c = idx & 63;
  float inv = rsqrtf(var[c] + 1e-5f);
  float u = g2[c]*(upd[idx] - mean[c])*inv + bt2[c];
  float x = fea[idx] + u;
  fea[idx] = fmaxf(x, 0.0f) + log1pf(expf(-fabsf(x)));
}

// ---------------- mean pool + lattice concat -> crys[G,73] ---------------------------
__global__ void cgc_pool(const float* __restrict__ fea, const float* __restrict__ lat,
                         float* __restrict__ crys) {
  int idx = blockIdx.x * blockDim.x + threadIdx.x;   // G*73 threads
  if (idx >= GNUM*73) return;
  int g = idx / 73, c = idx % 73;
  if (c < AF) {
    float s = 0.0f;
    for (int a = 0; a < APG; ++a) s += fea[(g*APG + a)*AF + c];
    crys[idx] = s * (1.0f / (float)APG);
  } else {
    crys[idx] = lat[g*9 + (c - AF)];
  }
}

// ---------------- fc1: h = silu(crys @ W1 + b1), [G,128] -----------------------------
__global__ void cgc_fc1(const float* __restrict__ crys, const float* __restrict__ W1,
                        const float* __restrict__ b1, float* __restrict__ h) {
  int idx = blockIdx.x * blockDim.x + threadIdx.x;   // G*128 threads
  int g = idx >> 7, j = idx & 127;
  float s = b1[j];
  for (int k = 0; k < 73; ++k) s += crys[g*73 + k] * W1[k*128 + j];
  h[idx] = s / (1.0f + expf(-s));                    // silu
}

// ---------------- fc2: out = h @ W2 + b2; split -> mu, logvar ------------------------
__global__ void cgc_fc2(const float* __restrict__ h, const float* __restrict__ W2,
                        const float* __restrict__ b2, float* __restrict__ out) {
  int idx = blockIdx.x * blockDim.x + threadIdx.x;   // G*256 threads
  int g = idx >> 8, j = idx & 255;
  float s = b2[j];
  for (int k = 0; k < 128; ++k) s += h[g*128 + k] * W2[k*256 + j];
  if (j < LD) out[g*LD + j] = s;                     // mu
  else        out[GNUM*LD + g*LD + (j - LD)] = s;    // logvar
}

extern "C" void kernel_launch(void* const* d_in, const int* in_sizes, int n_in,
                              void* d_out, int out_size, void* d_ws, size_t ws_size,
                              hipStream_t stream) {
  const float* lattice = (const float*)d_in[0];
  const float* fracs   = (const float*)d_in[1];
  const int*   species = (const int*)  d_in[2];
  // d_in[3] = batch_indices (structure is known: n/32)
  const float* emb     = (const float*)d_in[4];
  const float* W_full  = (const float*)d_in[5];
  const float* b_full  = (const float*)d_in[6];
  const float* g1      = (const float*)d_in[7];
  const float* bt1     = (const float*)d_in[8];
  const float* g2      = (const float*)d_in[9];
  const float* bt2     = (const float*)d_in[10];
  const float* W_fc1   = (const float*)d_in[11];
  const float* b_fc1   = (const float*)d_in[12];
  const float* W_fc2   = (const float*)d_in[13];
  const float* b_fc2   = (const float*)d_in[14];
  float* out = (float*)d_out;

  char* ws = (char*)d_ws;
  size_t off = 0;
  auto alloc = [&](size_t bytes) -> void* {
    void* p = ws + off;
    off += (bytes + 255) & ~(size_t)255;
    return p;
  };
  float* cart  = (float*)alloc((size_t)NATOMS*3*4);
  int*   esrc  = (int*)  alloc((size_t)EDGES*4);
  float* edist = (float*)alloc((size_t)EDGES*4);
  float* fea   = (float*)alloc((size_t)NATOMS*AF*4);
  float* total = (float*)alloc((size_t)EDGES*C2*4);
  float* mean1 = (float*)alloc(C2*4);
  float* var1  = (float*)alloc(C2*4);
  float* upd   = (float*)alloc((size_t)NATOMS*AF*4);
  float* mean2 = (float*)alloc(AF*4);
  float* var2  = (float*)alloc(AF*4);
  float* crys  = (float*)alloc((size_t)GNUM*73*4);
  float* hbuf  = (float*)alloc((size_t)GNUM*128*4);

  cgc_cart<<<(NATOMS+255)/256, 256, 0, stream>>>(fracs, lattice, cart);
  cgc_knn<<<GNUM, APG, 0, stream>>>(cart, esrc, edist);
  cgc_fea_init<<<(NATOMS*AF)/256, 256, 0, stream>>>(species, emb, fea);

  const size_t wt_lds = (size_t)C2 * WSTRIDE * sizeof(float);  // ~97 KB
  for (int l = 0; l < 3; ++l) {
    cgc_edge_gemm<<<EDGES/64, 128, wt_lds, stream>>>(
        fea, esrc, edist, W_full + (size_t)l*FIN*C2, b_full + l*C2, total);
    cgc_col_stats<<<C2, 256, 0, stream>>>(total, EDGES, C2, mean1, var1);
    cgc_msg_seg<<<(NATOMS*AF)/256, 256, 0, stream>>>(
        total, mean1, var1, g1 + l*C2, bt1 + l*C2, upd);
    cgc_col_stats<<<AF, 256, 0, stream>>>(upd, NATOMS, AF, mean2, var2);
    cgc_fea_update<<<(NATOMS*AF)/256, 256, 0, stream>>>(
        fea, upd, mean2, var2, g2 + l*AF, bt2 + l*AF);
  }

  cgc_pool<<<(GNUM*73 + 255)/256, 256, 0, stream>>>(fea, lattice, crys);
  cgc_fc1<<<(GNUM*128)/256, 256, 0, stream>>>(crys, W_fc1, b_fc1, hbuf);
  cgc_fc2<<<(GNUM*256)/256, 256, 0, stream>>>(hbuf, W_fc2, b_fc2, out);
}